// ChebConv_64080912056891
// MI455X (gfx1250) — compile-verified
//
#include <hip/hip_runtime.h>

// Problem constants (from reference)
#define B_   4
#define N_   512
#define C_   32
#define K1_  4
#define H_   4
#define P_   64
#define ACT_ 256
#define KC_  (K1_*C_)   // 128

typedef float v2f __attribute__((ext_vector_type(2)));
typedef float v8f __attribute__((ext_vector_type(8)));

static __device__ __forceinline__ v8f wmma_f32(v2f a, v2f b, v8f c) {
  // V_WMMA_F32_16X16X4_F32 : A 16x4 f32 (2 VGPR), B 4x16 f32 (2 VGPR), C/D 16x16 f32 (8 VGPR)
  // 8-arg pattern: (neg_a, A, neg_b, B, c_mod, C, reuse_a, reuse_b)
  return __builtin_amdgcn_wmma_f32_16x16x4_f32(false, a, false, b, (short)0, c, false, false);
}

// ---------------------------------------------------------------------------
// Kernel 1: attention scores si/sj (B,N,H) for real & imag
// ---------------------------------------------------------------------------
__global__ void prep_scores(const float* __restrict__ Xr, const float* __restrict__ Xi,
                            const float* __restrict__ AWr, const float* __restrict__ AWi,
                            float* __restrict__ si_re, float* __restrict__ si_im,
                            float* __restrict__ sj_re, float* __restrict__ sj_im)
{
  int idx = blockIdx.x * blockDim.x + threadIdx.x;
  if (idx >= B_ * N_) return;
  const float* xr = Xr + (size_t)idx * C_;
  const float* xi = Xi + (size_t)idx * C_;
  float sir[H_] = {0,0,0,0}, sii[H_] = {0,0,0,0};
  float sjr[H_] = {0,0,0,0}, sji[H_] = {0,0,0,0};
  for (int c = 0; c < C_; ++c) {
    float a = xr[c], b = xi[c];
#pragma unroll
    for (int h = 0; h < H_; ++h) {
      float w1r = AWr[c * H_ + h],        w1i = AWi[c * H_ + h];
      float w2r = AWr[(C_ + c) * H_ + h], w2i = AWi[(C_ + c) * H_ + h];
      sir[h] += a * w1r - b * w1i;
      sii[h] += a * w1i + b * w1r;
      sjr[h] += a * w2r - b * w2i;
      sji[h] += a * w2i + b * w2r;
    }
  }
#pragma unroll
  for (int h = 0; h < H_; ++h) {
    si_re[(size_t)idx * H_ + h] = sir[h];
    si_im[(size_t)idx * H_ + h] = sii[h];
    sj_re[(size_t)idx * H_ + h] = sjr[h];
    sj_im[(size_t)idx * H_ + h] = sji[h];
  }
}

// ---------------------------------------------------------------------------
// Kernel 2: XT[b][c64][j]  (c64 in [0,64): 0..31 = Xr channel, 32..63 = Xi channel)
// ---------------------------------------------------------------------------
__global__ void prep_xt(const float* __restrict__ Xr, const float* __restrict__ Xi,
                        float* __restrict__ XT)
{
  int idx = blockIdx.x * blockDim.x + threadIdx.x;
  if (idx >= B_ * 64 * N_) return;
  int j = idx % N_;
  int rem = idx / N_;
  int c = rem % 64;
  int b = rem / 64;
  float v = (c < C_) ? Xr[((size_t)b * N_ + j) * C_ + c]
                     : Xi[((size_t)b * N_ + j) * C_ + (c - C_)];
  XT[idx] = v;
}

// ---------------------------------------------------------------------------
// Kernel 3: repack weights: WT[h][k*32+c][p] = W[k][c][p*H+h]
// ---------------------------------------------------------------------------
__global__ void prep_wt(const float* __restrict__ Wr, const float* __restrict__ Wi,
                        float* __restrict__ WTr, float* __restrict__ WTi)
{
  int idx = blockIdx.x * blockDim.x + threadIdx.x;
  if (idx >= H_ * KC_ * P_) return;
  int p = idx % P_;
  int rem = idx / P_;
  int kc = rem % KC_;
  int h = rem / KC_;
  int k = kc / C_, c = kc % C_;
  size_t src = ((size_t)k * C_ + c) * ACT_ + (size_t)p * H_ + h;
  WTr[idx] = Wr[src];
  WTi[idx] = Wi[src];
}

// ---------------------------------------------------------------------------
// Kernel 4: masked squared-PReLU scores + softmax over j.
// attn layout: [(b*H+h)][n][j]   (row-major, 512x512 per (b,h))
// One 256-thread block per (b,n); each thread covers j = t and t+256.
// ---------------------------------------------------------------------------
__global__ void attn_softmax(const float* __restrict__ Lr,
                             const float* __restrict__ si_re, const float* __restrict__ si_im,
                             const float* __restrict__ sj_re, const float* __restrict__ sj_im,
                             const float* __restrict__ bre, const float* __restrict__ bim,
                             const float* __restrict__ pa_re, const float* __restrict__ pa_im,
                             float* __restrict__ attn)
{
  __shared__ float red[256];
  int b = blockIdx.x / N_;
  int n = blockIdx.x % N_;
  int t = threadIdx.x;
  int j0 = t, j1 = t + 256;

  // mask[b,n,j] = sum_k |L_real[b,k,n,j]| > 1e-9
  float m0 = 0.f, m1 = 0.f;
#pragma unroll
  for (int k = 0; k < K1_; ++k) {
    const float* lp = Lr + (((size_t)(b * K1_ + k) * N_) + n) * N_;
    m0 += fabsf(lp[j0]);
    m1 += fabsf(lp[j1]);
  }
  bool mask0 = m0 > 1e-9f;
  bool mask1 = m1 > 1e-9f;

  const float a_re = pa_re[0], a_im = pa_im[0];
  const float NEG = -1.0e9f;

  for (int h = 0; h < H_; ++h) {
    float sre = si_re[((size_t)b * N_ + n) * H_ + h];
    float sim = si_im[((size_t)b * N_ + n) * H_ + h];
    float br = bre[h], bi = bim[h];

    // score at j0
    float xr0 = sre + sj_re[((size_t)b * N_ + j0) * H_ + h] + br;
    xr0 = (xr0 >= 0.f) ? xr0 : a_re * xr0;
    float xi0 = sim + sj_im[((size_t)b * N_ + j0) * H_ + h] + bi;
    xi0 = (xi0 >= 0.f) ? xi0 : a_im * xi0;
    float rm0 = mask0 ? xr0 : NEG;
    float im0 = mask0 ? xi0 : NEG;
    float sc0 = rm0 * rm0 + im0 * im0;
    // score at j1
    float xr1 = sre + sj_re[((size_t)b * N_ + j1) * H_ + h] + br;
    xr1 = (xr1 >= 0.f) ? xr1 : a_re * xr1;
    float xi1 = sim + sj_im[((size_t)b * N_ + j1) * H_ + h] + bi;
    xi1 = (xi1 >= 0.f) ? xi1 : a_im * xi1;
    float rm1 = mask1 ? xr1 : NEG;
    float im1 = mask1 ? xi1 : NEG;
    float sc1 = rm1 * rm1 + im1 * im1;

    // block max
    red[t] = fmaxf(sc0, sc1);
    __syncthreads();
    for (int s = 128; s >= 1; s >>= 1) {
      if (t < s) red[t] = fmaxf(red[t], red[t + s]);
      __syncthreads();
    }
    float mx = red[0];
    __syncthreads();

    float e0 = __expf(sc0 - mx);
    float e1 = __expf(sc1 - mx);

    // block sum
    red[t] = e0 + e1;
    __syncthreads();
    for (int s = 128; s >= 1; s >>= 1) {
      if (t < s) red[t] += red[t + s];
      __syncthreads();
    }
    float inv = 1.0f / red[0];
    __syncthreads();

    float* ap = attn + (((size_t)(b * H_ + h) * N_) + n) * N_;
    ap[j0] = e0 * inv;
    ap[j1] = e1 * inv;
  }
}

// ---------------------------------------------------------------------------
// Kernel 5: main einsum via fp32 WMMA.
// Per wave: 16 output rows of (b,k,h); D = (L.*attn) @ XT^T over j (K=512).
// accR[t] = Mr@Xcat tile t, accI[t] = Mi@Xcat tile t (t: cols 0-15,16-31 -> Xr; 32-47,48-63 -> Xi)
// LXr = Mr@Xr - Mi@Xi ; LXi = Mr@Xi + Mi@Xr
// Output layout: LX[(b*H+h)][n][k*32+c]
// ---------------------------------------------------------------------------
__global__ __launch_bounds__(128) void cheb_spmm_wmma(
    const float* __restrict__ Lr, const float* __restrict__ Li,
    const float* __restrict__ attn, const float* __restrict__ XT,
    float* __restrict__ LXr, float* __restrict__ LXi)
{
  int bid = blockIdx.x;
  int nblk = bid & 7;  int rem = bid >> 3;
  int h = rem & 3;     rem >>= 2;
  int k = rem & 3;     int b = rem >> 2;
  int wave = threadIdx.x >> 5;
  int lane = threadIdx.x & 31;
  int n_base = nblk * 64 + wave * 16;

  // fp32 A-matrix 16x4 layout: lanes 0-15 -> K=0,1 ; lanes 16-31 -> K=2,3 (rows M=lane&15)
  int r  = lane & 15;
  int kb = (lane >> 4) << 1;

  const float* lrp = Lr   + (((size_t)(b * K1_ + k) * N_) + (n_base + r)) * N_ + kb;
  const float* lip = Li   + (((size_t)(b * K1_ + k) * N_) + (n_base + r)) * N_ + kb;
  const float* atp = attn + (((size_t)(b * H_  + h) * N_) + (n_base + r)) * N_ + kb;
  const float* xtp = XT   + (size_t)b * 64 * N_ + kb;   // + c*N_ + j0

  const v8f zero = {0.f,0.f,0.f,0.f,0.f,0.f,0.f,0.f};
  v8f accR[4] = {zero, zero, zero, zero};
  v8f accI[4] = {zero, zero, zero, zero};

  for (int j0 = 0; j0 < N_; j0 += 4) {
    // prefetch next cacheline of the streaming rows
    __builtin_prefetch(lrp + j0 + 16, 0, 1);
    __builtin_prefetch(lip + j0 + 16, 0, 1);
    __builtin_prefetch(atp + j0 + 16, 0, 1);

    v2f lr = *(const v2f*)(lrp + j0);
    v2f li = *(const v2f*)(lip + j0);
    v2f at = *(const v2f*)(atp + j0);
    v2f Ar = lr * at;
    v2f Ai = li * at;
#pragma unroll
    for (int t = 0; t < 4; ++t) {
      v2f Bt = *(const v2f*)(xtp + (size_t)(t * 16 + r) * N_ + j0);
      accR[t] = wmma_f32(Ar, Bt, accR[t]);
      accI[t] = wmma_f32(Ai, Bt, accI[t]);
    }
  }

  // D layout: VGPR v -> row v (+8 for lanes>=16); col = lane&15
  int rowoff = (lane >> 4) * 8;
  int cl = lane & 15;
#pragma unroll
  for (int t2 = 0; t2 < 2; ++t2) {
#pragma unroll
    for (int v = 0; v < 8; ++v) {
      int n = n_base + rowoff + v;
      int kc = k * C_ + t2 * 16 + cl;
      size_t o = (((size_t)(b * H_ + h) * N_) + n) * KC_ + kc;
      LXr[o] = accR[t2][v] - accI[t2 + 2][v];
      LXi[o] = accR[t2 + 2][v] + accI[t2][v];
    }
  }
}

// ---------------------------------------------------------------------------
// Kernel 6: projection via fp32 WMMA.
// Per wave: 16 rows of (b,h): Y[n,p] over kc (K=128), 64 p columns.
// Yre = LXr@Wr - LXi@Wi ; Yim = LXr@Wi + LXi@Wr
// out[b][n][p*H+h] (+bias), imag half offset by B*N*ACT
// ---------------------------------------------------------------------------
__global__ __launch_bounds__(128) void proj_wmma(
    const float* __restrict__ LXr, const float* __restrict__ LXi,
    const float* __restrict__ WTr, const float* __restrict__ WTi,
    const float* __restrict__ bre, const float* __restrict__ bim,
    float* __restrict__ out)
{
  int bid = blockIdx.x;
  int nblk = bid & 7; int rem = bid >> 3;
  int h = rem & 3;    int b = rem >> 2;
  int wave = threadIdx.x >> 5;
  int lane = threadIdx.x & 31;
  int n_base = nblk * 64 + wave * 16;
  int r  = lane & 15;
  int kb = (lane >> 4) << 1;

  const float* arp = LXr + (((size_t)(b * H_ + h) * N_) + (n_base + r)) * KC_ + kb;
  const float* aip = LXi + (((size_t)(b * H_ + h) * N_) + (n_base + r)) * KC_ + kb;
  const float* wrp = WTr + (size_t)h * KC_ * P_ + (size_t)kb * P_ + r;  // + kc0*P_ + t*16
  const float* wip = WTi + (size_t)h * KC_ * P_ + (size_t)kb * P_ + r;

  const v8f zero = {0.f,0.f,0.f,0.f,0.f,0.f,0.f,0.f};
  v8f aRe[4] = {zero, zero, zero, zero};
  v8f aIm[4] = {zero, zero, zero, zero};

  for (int kc0 = 0; kc0 < KC_; kc0 += 4) {
    v2f ar = *(const v2f*)(arp + kc0);
    v2f ai = *(const v2f*)(aip + kc0);
    v2f nai = -ai;
#pragma unroll
    for (int t = 0; t < 4; ++t) {
      const float* w0 = wrp + (size_t)kc0 * P_ + t * 16;
      const float* w1 = wip + (size_t)kc0 * P_ + t * 16;
      v2f Br = {w0[0], w0[P_]};
      v2f Bi = {w1[0], w1[P_]};
      aRe[t] = wmma_f32(ar,  Br, aRe[t]);
      aRe[t] = wmma_f32(nai, Bi, aRe[t]);
      aIm[t] = wmma_f32(ar,  Bi, aIm[t]);
      aIm[t] = wmma_f32(ai,  Br, aIm[t]);
    }
  }

  int rowoff = (lane >> 4) * 8;
  int cl = lane & 15;
  const size_t half = (size_t)B_ * N_ * ACT_;
#pragma unroll
  for (int t = 0; t < 4; ++t) {
#pragma unroll
    for (int v = 0; v < 8; ++v) {
      int n = n_base + rowoff + v;
      int p = t * 16 + cl;
      size_t o = ((size_t)b * N_ + n) * ACT_ + (size_t)p * H_ + h;
      out[o]        = aRe[t][v] + bre[p * H_ + h];
      out[half + o] = aIm[t][v] + bim[p * H_ + h];
    }
  }
}

// ---------------------------------------------------------------------------
// Host launcher
// ---------------------------------------------------------------------------
extern "C" void kernel_launch(void* const* d_in, const int* in_sizes, int n_in,
                              void* d_out, int out_size, void* d_ws, size_t ws_size,
                              hipStream_t stream)
{
  (void)in_sizes; (void)n_in; (void)out_size;
  const float* Xr  = (const float*)d_in[0];
  const float* Xi  = (const float*)d_in[1];
  const float* Lr  = (const float*)d_in[2];
  const float* Li  = (const float*)d_in[3];
  const float* Wr  = (const float*)d_in[4];
  const float* Wi  = (const float*)d_in[5];
  const float* AWr = (const float*)d_in[6];
  const float* AWi = (const float*)d_in[7];
  const float* ABr = (const float*)d_in[8];
  const float* ABi = (const float*)d_in[9];
  const float* PAr = (const float*)d_in[10];
  const float* PAi = (const float*)d_in[11];
  const float* BRr = (const float*)d_in[12];
  const float* BRi = (const float*)d_in[13];
  float* out = (float*)d_out;
  float* ws  = (float*)d_ws;

  // workspace layout (floats); total = 6,520,832 floats (~26 MB)
  const size_t SI_RE = 0;
  const size_t SI_IM = SI_RE + (size_t)B_ * N_ * H_;       // 8192
  const size_t SJ_RE = SI_IM + (size_t)B_ * N_ * H_;
  const size_t SJ_IM = SJ_RE + (size_t)B_ * N_ * H_;
  const size_t XT_O  = SJ_IM + (size_t)B_ * N_ * H_;       // 32768
  const size_t WTR_O = XT_O  + (size_t)B_ * 64 * N_;       // +131072
  const size_t WTI_O = WTR_O + (size_t)H_ * KC_ * P_;      // +32768
  const size_t ATT_O = WTI_O + (size_t)H_ * KC_ * P_;      // +32768
  const size_t LXR_O = ATT_O + (size_t)B_ * H_ * N_ * N_;  // +4194304
  const size_t LXI_O = LXR_O + (size_t)B_ * H_ * N_ * KC_; // +1048576
  (void)ws_size; // requires >= (LXI_O + B_*H_*N_*KC_)*4 bytes (~26 MB)

  prep_scores<<<(B_ * N_ + 255) / 256, 256, 0, stream>>>(
      Xr, Xi, AWr, AWi, ws + SI_RE, ws + SI_IM, ws + SJ_RE, ws + SJ_IM);

  prep_xt<<<(B_ * 64 * N_ + 255) / 256, 256, 0, stream>>>(Xr, Xi, ws + XT_O);

  prep_wt<<<(H_ * KC_ * P_ + 255) / 256, 256, 0, stream>>>(Wr, Wi, ws + WTR_O, ws + WTI_O);

  attn_softmax<<<B_ * N_, 256, 0, stream>>>(
      Lr, ws + SI_RE, ws + SI_IM, ws + SJ_RE, ws + SJ_IM,
      ABr, ABi, PAr, PAi, ws + ATT_O);

  cheb_spmm_wmma<<<B_ * K1_ * H_ * (N_ / 64), 128, 0, stream>>>(
      Lr, Li, ws + ATT_O, ws + XT_O, ws + LXR_O, ws + LXI_O);

  proj_wmma<<<B_ * H_ * (N_ / 64), 128, 0, stream>>>(
      ws + LXR_O, ws + LXI_O, ws + WTR_O, ws + WTI_O, BRr, BRi, out);
}